// lr_nufft_48704929136778
// MI455X (gfx1250) — compile-verified
//
#include <hip/hip_runtime.h>
#include <math.h>

// ---------------------------------------------------------------------------
// NUFFT forward for MI455X (gfx1250):
//   stage 0: build fused centered-DFT operator M (320x256 complex)
//            M[k][j] = apod[j] * exp(-2*pi*i*(k-160)*(j-128)/320)
//            (folds apodization, zero-pad offset, ifftshift/fftshift)
//   stage 1: Y[c]   = M @ img[c]        (complex GEMM, WMMA f32 16x16x4)
//   stage 2: K[c]   = Y[c] @ M^T        (complex GEMM, WMMA f32 16x16x4)
//   stage 3: 6x6 Kaiser-Bessel gather at 262144 points, 8 channels
// ---------------------------------------------------------------------------

typedef __attribute__((ext_vector_type(2))) float v2f;
typedef __attribute__((ext_vector_type(8))) float v8f;

#define NIMG   256
#define NOS    320
#define NCH    8
#define NPTS   (64 * 4096)
#define KBW    6.0f

// ------------------------- stage 0: DFT operator ---------------------------
__global__ void build_dft_op(float* __restrict__ Mr, float* __restrict__ Mi) {
    int k = blockIdx.x;        // 0..319  (output freq index)
    int j = threadIdx.x;       // 0..255  (input spatial index)
    const float PI = 3.14159265358979323846f;
    const float BETA = PI * sqrtf((KBW / 1.25f * (1.25f - 0.5f)) * (KBW / 1.25f * (1.25f - 0.5f)) - 0.8f);
    // apodization (inverse FT of KB kernel), axis length 256, os length 320
    float t = PI * KBW * (float)(j - NIMG / 2) / (float)NOS;
    float arg = sqrtf(fmaxf(BETA * BETA - t * t, 0.0f));
    float apod = arg / (0.5f * (expf(arg) - expf(-arg)));   // arg/sinh(arg)
    // centered DFT phase in double (product up to ~2e4 -> needs wide reduction)
    double ang = -2.0 * 3.141592653589793238462643 *
                 (double)((k - NOS / 2) * (j - NIMG / 2)) / (double)NOS;
    Mr[k * NIMG + j] = apod * (float)cos(ang);
    Mi[k * NIMG + j] = apod * (float)sin(ang);
}

// ---------------- stages 1/2: complex GEMM on the WMMA pipe ----------------
// D = A(MxK complex) * B(KxN complex), one wave per 16x16 output tile.
// B_TRANS==false : B element (k,n) at Br[k*ldb + n]
// B_TRANS==true  : B element (k,n) at Br[n*ldb + k]  (B = M^T, M row-major)
template <bool B_TRANS>
__global__ void cgemm_wmma_f32(const float* __restrict__ Ar, const float* __restrict__ Ai,
                               int lda, int a_bstride,
                               const float* __restrict__ Br, const float* __restrict__ Bi,
                               int ldb, int b_bstride,
                               float* __restrict__ Cr, float* __restrict__ Ci,
                               int ldc, int c_bstride, int Kdim) {
    const int lane  = threadIdx.x;                 // 0..31, full wave (EXEC all ones)
    const int tileM = blockIdx.x * 16;
    const int tileN = blockIdx.y * 16;
    const int batch = blockIdx.z;

    Ar += (size_t)batch * a_bstride;  Ai += (size_t)batch * a_bstride;
    Br += (size_t)batch * b_bstride;  Bi += (size_t)batch * b_bstride;
    Cr += (size_t)batch * c_bstride;  Ci += (size_t)batch * c_bstride;

    // A fragment: lanes 0-15 hold K=k+0,k+1 for row tileM+lane;
    //             lanes 16-31 hold K=k+2,k+3 for row tileM+lane-16
    const int a_row = tileM + (lane & 15);
    const int a_k0  = (lane < 16) ? 0 : 2;
    // B fragment: same half-wave K split, N = tileN + lane&15
    const int b_col = tileN + (lane & 15);
    const int b_k0  = (lane < 16) ? 0 : 2;

    v8f p_rr = {}; v8f p_ii = {}; v8f p_ri = {}; v8f p_ir = {};

#pragma unroll 4
    for (int k = 0; k < Kdim; k += 4) {
        // prefetch one unrolled macro-tile (16 K-steps) ahead; operands are
        // L2-resident, this pulls lines toward the WGP (global_prefetch_b8)
        __builtin_prefetch(&Ar[a_row * lda + k + 16], 0, 1);
        __builtin_prefetch(&Ai[a_row * lda + k + 16], 0, 1);

        v2f ar, ai, br, bi;
        const int abase = a_row * lda + k + a_k0;
        ar.x = Ar[abase];     ar.y = Ar[abase + 1];
        ai.x = Ai[abase];     ai.y = Ai[abase + 1];
        if (!B_TRANS) {
            const int b0 = (k + b_k0) * ldb + b_col;
            br.x = Br[b0];            br.y = Br[b0 + ldb];
            bi.x = Bi[b0];            bi.y = Bi[b0 + ldb];
        } else {
            const int b0 = b_col * ldb + k + b_k0;
            br.x = Br[b0];            br.y = Br[b0 + 1];
            bi.x = Bi[b0];            bi.y = Bi[b0 + 1];
        }
        // D = A*B + C on the matrix pipe (fp32, full precision)
        p_rr = __builtin_amdgcn_wmma_f32_16x16x4_f32(false, ar, false, br, (short)0, p_rr, false, false);
        p_ii = __builtin_amdgcn_wmma_f32_16x16x4_f32(false, ai, false, bi, (short)0, p_ii, false, false);
        p_ri = __builtin_amdgcn_wmma_f32_16x16x4_f32(false, ar, false, bi, (short)0, p_ri, false, false);
        p_ir = __builtin_amdgcn_wmma_f32_16x16x4_f32(false, ai, false, br, (short)0, p_ir, false, false);
    }

    // C/D layout: VGPR r -> row tileM + r + (lane>=16 ? 8 : 0), col tileN + lane&15
    const int n      = tileN + (lane & 15);
    const int m_base = tileM + ((lane < 16) ? 0 : 8);
#pragma unroll
    for (int r = 0; r < 8; ++r) {
        const int m = m_base + r;
        Cr[m * ldc + n] = p_rr[r] - p_ii[r];   // Re = ArBr - AiBi
        Ci[m * ldc + n] = p_ri[r] + p_ir[r];   // Im = ArBi + AiBr
    }
}

// ------------------------- stage 3: KB interpolation -----------------------
__device__ __forceinline__ float bessel_i0(float x) {
    // Abramowitz & Stegun 9.8.1 / 9.8.2
    float ax = fabsf(x);
    if (ax < 3.75f) {
        float t = x / 3.75f; t *= t;
        return 1.0f + t * (3.5156229f + t * (3.0899424f + t * (1.2067492f +
                     t * (0.2659732f + t * (0.0360768f + t * 0.0045813f)))));
    }
    float t = 3.75f / ax;
    return (expf(ax) * __frsqrt_rn(ax)) *
           (0.39894228f + t * (0.01328592f + t * (0.00225319f + t * (-0.00157565f +
            t * (0.00916281f + t * (-0.02057706f + t * (0.02635537f +
            t * (-0.01647633f + t * 0.00392377f))))))));
}

__global__ void kb_interp(const float* __restrict__ trj,
                          const float* __restrict__ Kr, const float* __restrict__ Ki,
                          float* __restrict__ out) {
    int p = blockIdx.x * blockDim.x + threadIdx.x;
    if (p >= NPTS) return;
    const float PI = 3.14159265358979323846f;
    const float BETA = PI * sqrtf((KBW / 1.25f * (1.25f - 0.5f)) * (KBW / 1.25f * (1.25f - 0.5f)) - 0.8f);

    float cx = trj[2 * p + 0] * 1.25f + 160.0f;   // axis 0 (H)
    float cy = trj[2 * p + 1] * 1.25f + 160.0f;   // axis 1 (W)
    float x0 = ceilf(cx - KBW * 0.5f);
    float y0 = ceilf(cy - KBW * 0.5f);

    float wx[6], wy[6];
    int   ix[6], iy[6];
#pragma unroll
    for (int d = 0; d < 6; ++d) {
        float px = x0 + (float)d;
        float u  = (px - cx) / (KBW * 0.5f);
        wx[d] = bessel_i0(BETA * sqrtf(fmaxf(1.0f - u * u, 0.0f)));
        int i = (int)px % NOS; if (i < 0) i += NOS;
        ix[d] = i;

        float py = y0 + (float)d;
        float v  = (py - cy) / (KBW * 0.5f);
        wy[d] = bessel_i0(BETA * sqrtf(fmaxf(1.0f - v * v, 0.0f)));
        int j = (int)py % NOS; if (j < 0) j += NOS;
        iy[d] = j;
    }

    float accr[NCH], acci[NCH];
#pragma unroll
    for (int c = 0; c < NCH; ++c) { accr[c] = 0.0f; acci[c] = 0.0f; }

#pragma unroll
    for (int dx = 0; dx < 6; ++dx) {
        const int rowb = ix[dx] * NOS;
#pragma unroll
        for (int dy = 0; dy < 6; ++dy) {
            const float w    = wx[dx] * wy[dy];
            const int   base = rowb + iy[dy];
#pragma unroll
            for (int c = 0; c < NCH; ++c) {
                accr[c] = fmaf(Kr[c * (NOS * NOS) + base], w, accr[c]);
                acci[c] = fmaf(Ki[c * (NOS * NOS) + base], w, acci[c]);
            }
        }
    }

    // fold scales: 1/sqrt(256*256) from FFT norm, 1/36 from kernel width^2
    const float s = 1.0f / (36.0f * 256.0f);
#pragma unroll
    for (int c = 0; c < NCH; ++c) {
        out[2 * (c * NPTS + p) + 0] = accr[c] * s;
        out[2 * (c * NPTS + p) + 1] = acci[c] * s;
    }
}

// ------------------------------- launcher ----------------------------------
extern "C" void kernel_launch(void* const* d_in, const int* in_sizes, int n_in,
                              void* d_out, int out_size, void* d_ws, size_t ws_size,
                              hipStream_t stream) {
    (void)in_sizes; (void)n_in; (void)out_size; (void)ws_size;
    const float* img_r = (const float*)d_in[0];  // (1,8,256,256)
    const float* img_i = (const float*)d_in[1];  // (1,8,256,256)
    const float* trj   = (const float*)d_in[2];  // (1,64,4096,2)
    float* out = (float*)d_out;                  // (1,8,64,4096) complex64 -> interleaved

    // workspace layout (floats), everything written before read
    float* ws = (float*)d_ws;
    float* Mr = ws;                                  // 320*256
    float* Mi = Mr + NOS * NIMG;                     // 320*256
    float* Yr = Mi + NOS * NIMG;                     // 8*320*256
    float* Yi = Yr + NCH * NOS * NIMG;
    float* Kr = Yi + NCH * NOS * NIMG;               // 8*320*320
    float* Ki = Kr + NCH * NOS * NOS;

    // stage 0: fused centered-DFT operator (runs once, tiny)
    build_dft_op<<<dim3(NOS), dim3(NIMG), 0, stream>>>(Mr, Mi);

    // stage 1: Y[c] = M (320x256) @ img[c] (256x256);  one wave per 16x16 tile
    cgemm_wmma_f32<false><<<dim3(NOS / 16, NIMG / 16, NCH), dim3(32), 0, stream>>>(
        Mr, Mi, NIMG, 0,
        img_r, img_i, NIMG, NIMG * NIMG,
        Yr, Yi, NIMG, NOS * NIMG, /*K=*/NIMG);

    // stage 2: K[c] = Y[c] (320x256) @ M^T (256x320)
    cgemm_wmma_f32<true><<<dim3(NOS / 16, NOS / 16, NCH), dim3(32), 0, stream>>>(
        Yr, Yi, NIMG, NOS * NIMG,
        Mr, Mi, NIMG, 0,
        Kr, Ki, NOS, NOS * NOS, /*K=*/NIMG);

    // stage 3: 6x6 KB gather out of L2-resident k-space
    kb_interp<<<dim3((NPTS + 255) / 256), dim3(256), 0, stream>>>(trj, Kr, Ki, out);
}